// FGMSA_58720792871402
// MI455X (gfx1250) — compile-verified
//
#include <hip/hip_runtime.h>
#include <math.h>

typedef __attribute__((ext_vector_type(2))) float v2f;
typedef __attribute__((ext_vector_type(8))) float v8f;

#define B_   2
#define H_   32
#define W_   32
#define NH_  12
#define DH_  32
#define NG_  6
#define NC_  384
#define GC_  64
#define NS_  1024      // H*W
#define TBL_ 63        // 2H-1

// ---------------------------------------------------------------------------
// Generic fp32 WMMA GEMM:  C[M,N] = A[M,K] @ W[K,N] + bias[N]
// One wave computes a 16x64 strip (4 accumulators share the A fragment).
// ---------------------------------------------------------------------------
__global__ void gemm_bias(const float* __restrict__ A, const float* __restrict__ Wt,
                          const float* __restrict__ bias, float* __restrict__ C,
                          int M, int N, int K) {
  const int lane = threadIdx.x & 31;
  const int wv   = threadIdx.x >> 5;
  const int gw   = blockIdx.x * (blockDim.x >> 5) + wv;
  const int nt4  = N >> 6;
  const int mt   = gw / nt4;
  const int nb   = (gw % nt4) * 64;
  if (mt * 16 >= M) return;
  const int hl = lane >> 4;
  const int r  = lane & 15;
  const int m0 = mt * 16;

  v8f acc0 = {}, acc1 = {}, acc2 = {}, acc3 = {};
  for (int k0 = 0; k0 < K; k0 += 4) {
    const int ka = k0 + 2 * hl;
    v2f a;
    a.x = A[(size_t)(m0 + r) * K + ka];
    a.y = A[(size_t)(m0 + r) * K + ka + 1];
    const float* w0 = Wt + (size_t)ka * N;
    const float* w1 = Wt + (size_t)(ka + 1) * N;
    v2f b0, b1, b2, b3;
    b0.x = w0[nb + r];      b0.y = w1[nb + r];
    b1.x = w0[nb + 16 + r]; b1.y = w1[nb + 16 + r];
    b2.x = w0[nb + 32 + r]; b2.y = w1[nb + 32 + r];
    b3.x = w0[nb + 48 + r]; b3.y = w1[nb + 48 + r];
    acc0 = __builtin_amdgcn_wmma_f32_16x16x4_f32(false, a, false, b0, (short)0, acc0, false, false);
    acc1 = __builtin_amdgcn_wmma_f32_16x16x4_f32(false, a, false, b1, (short)0, acc1, false, false);
    acc2 = __builtin_amdgcn_wmma_f32_16x16x4_f32(false, a, false, b2, (short)0, acc2, false, false);
    acc3 = __builtin_amdgcn_wmma_f32_16x16x4_f32(false, a, false, b3, (short)0, acc3, false, false);
  }
  #pragma unroll
  for (int rr = 0; rr < 8; ++rr) {
    const int row = m0 + rr + 8 * hl;
    float* cp = C + (size_t)row * N + nb + r;
    cp[0]  = acc0[rr] + bias[nb + r];
    cp[16] = acc1[rr] + bias[nb + 16 + r];
    cp[32] = acc2[rr] + bias[nb + 32 + r];
    cp[48] = acc3[rr] + bias[nb + 48 + r];
  }
}

// ---------------------------------------------------------------------------
// Grouped 3x3 conv (SAME, NG=6 groups of 64ch), HWIO weights, + bias.
// One thread per output element.
// ---------------------------------------------------------------------------
__global__ void conv_off(const float* __restrict__ q, const float* __restrict__ w,
                         const float* __restrict__ bofs, float* __restrict__ out) {
  const int idx = blockIdx.x * blockDim.x + threadIdx.x;
  if (idx >= B_ * NS_ * NC_) return;
  const int oc = idx % NC_;
  const int n  = (idx / NC_) % NS_;
  const int b  = idx / (NC_ * NS_);
  const int x  = n & 31, y = n >> 5;
  const int g  = oc >> 6;
  float acc = bofs[oc];
  for (int dy = 0; dy < 3; ++dy) {
    const int yy = y + dy - 1;
    if (yy < 0 || yy >= H_) continue;
    for (int dx = 0; dx < 3; ++dx) {
      const int xx = x + dx - 1;
      if (xx < 0 || xx >= W_) continue;
      const float* qp = q + ((size_t)(b * NS_ + yy * W_ + xx) * NC_) + g * GC_;
      const float* wp = w + ((size_t)(dy * 3 + dx) * GC_) * NC_ + oc;
      #pragma unroll 8
      for (int ic = 0; ic < GC_; ++ic)
        acc += qp[ic] * wp[(size_t)ic * NC_];
    }
  }
  out[idx] = acc;
}

// ---------------------------------------------------------------------------
// LayerNorm(384) + tanh-GELU + per-group 64->2 projection + tanh*16 + ref add.
// One wave per (b, n) row; writes pos & ref directly into the output tuple.
// ---------------------------------------------------------------------------
__global__ void ln_offset(const float* __restrict__ conv, const float* __restrict__ ln_g,
                          const float* __restrict__ ln_b, const float* __restrict__ wp,
                          float* __restrict__ pos_out, float* __restrict__ ref_out) {
  __shared__ float rowbuf[8][NC_];
  const int lane = threadIdx.x & 31;
  const int wv   = threadIdx.x >> 5;
  const int row  = blockIdx.x * 8 + wv;       // 0..2047
  const int b    = row >> 10;
  const int n    = row & 1023;
  const float* rp = conv + (size_t)row * NC_;

  float vals[12];
  float sum = 0.f, sq = 0.f;
  #pragma unroll
  for (int t = 0; t < 12; ++t) {
    float v = rp[lane + 32 * t];
    vals[t] = v; sum += v; sq += v * v;
  }
  #pragma unroll
  for (int off = 16; off; off >>= 1) {
    sum += __shfl_xor(sum, off, 32);
    sq  += __shfl_xor(sq,  off, 32);
  }
  const float mean = sum * (1.0f / NC_);
  const float var  = sq * (1.0f / NC_) - mean * mean;
  const float rstd = rsqrtf(var + 1e-3f);
  #pragma unroll
  for (int t = 0; t < 12; ++t) {
    const int c = lane + 32 * t;
    float v = (vals[t] - mean) * rstd * ln_g[c] + ln_b[c];
    float g = v * 0.5f * (1.0f + tanhf(0.7978845608028654f * (v + 0.044715f * v * v * v)));
    rowbuf[wv][c] = g;
  }
  __syncthreads();
  if (lane < 12) {
    const int g = lane >> 1, dim = lane & 1;
    float dot = 0.f;
    #pragma unroll 8
    for (int c = 0; c < GC_; ++c) dot += rowbuf[wv][g * GC_ + c] * wp[c * 2 + dim];
    const float off = tanhf(dot) * 16.0f;                 // H/2 == W/2 == 16
    const float ref = (dim == 0) ? (float)(n & 31) : (float)(n >> 5);
    const size_t oi = (((size_t)(b * NG_ + g) * NS_) + n) * 2 + dim;
    pos_out[oi] = off + ref;
    ref_out[oi] = ref;
  }
}

// ---------------------------------------------------------------------------
// Fused flash attention, fp32 WMMA.  One wave per (batch-head, 16-query tile).
//   S^T = K · Q^T  (rows=keys -> softmax over keys is per-lane + one SWAPX16)
//   bias(q,k)      bilinear gather from 63x63 RPE table at (i_q-pos1, j_q-pos0)
//   O^T += V^T · P^T  with P^T B-fragments rebuilt from C-layout via shfl_xor.
// ---------------------------------------------------------------------------
__global__ void attn_fused(const float* __restrict__ qbuf, const float* __restrict__ kbuf,
                           const float* __restrict__ vbuf, const float* __restrict__ posb,
                           const float* __restrict__ rpe,  float* __restrict__ outb) {
  const int lane = threadIdx.x & 31;
  const int wv   = threadIdx.x >> 5;
  const int gw   = blockIdx.x * (blockDim.x >> 5) + wv;    // 0..1535
  const int bh   = gw >> 6;                                // 0..23
  const int qt   = gw & 63;
  const int b    = bh / NH_;
  const int h    = bh % NH_;
  const int g    = h >> 1;
  const int hl   = lane >> 4;
  const int r    = lane & 15;

  const float* Q   = qbuf + (size_t)b * NS_ * NC_;
  const float* Kp  = kbuf + (size_t)b * NS_ * NC_;
  const float* Vp  = vbuf + (size_t)b * NS_ * NC_;
  const float* pos = posb + (size_t)(b * NG_ + g) * NS_ * 2;

  const int nq = qt * 16 + r;          // this lane's query column
  const int iq = nq >> 5, jq = nq & 31;

  // Q^T B-fragments, resident for the whole key loop (16 floats/lane)
  v2f qf[8];
  #pragma unroll
  for (int c = 0; c < 8; ++c) {
    const int d = 4 * c + 2 * hl;
    qf[c].x = Q[(size_t)nq * NC_ + h * DH_ + d];
    qf[c].y = Q[(size_t)nq * NC_ + h * DH_ + d + 1];
  }

  float mrow = -1e30f, lrow = 0.f;
  v8f o0 = {}, o1 = {};
  const float scale = 0.17677669529663687f;    // 32^-0.5

  for (int kt = 0; kt < 64; ++kt) {
    // ---- S^T tile: rows=keys, cols=queries ----
    v8f s = {};
    const int nk_a = kt * 16 + r;
    #pragma unroll
    for (int c = 0; c < 8; ++c) {
      const int d = 4 * c + 2 * hl;
      v2f a;
      a.x = Kp[(size_t)nk_a * NC_ + h * DH_ + d];
      a.y = Kp[(size_t)nk_a * NC_ + h * DH_ + d + 1];
      s = __builtin_amdgcn_wmma_f32_16x16x4_f32(false, a, false, qf[c], (short)0, s, false, false);
    }

    // ---- scale + bilinear RPE bias, tile max ----
    float p[8];
    float tmax = -1e30f;
    #pragma unroll
    for (int rr = 0; rr < 8; ++rr) {
      const int nk = kt * 16 + rr + 8 * hl;
      const float p0 = pos[nk * 2 + 0];
      const float p1 = pos[nk * 2 + 1];
      const float xw = (float)iq - p1;
      const float yw = (float)jq - p0;
      const float x0f = floorf(xw), y0f = floorf(yw);
      const int x0 = (int)x0f, y0 = (int)y0f;
      const float wx = xw - x0f, wy = yw - y0f;
      float v00 = 0.f, v01 = 0.f, v10 = 0.f, v11 = 0.f;
      if ((unsigned)y0 < TBL_ && (unsigned)x0 < TBL_)             v00 = rpe[(y0 * TBL_ + x0) * NH_ + h];
      if ((unsigned)y0 < TBL_ && (unsigned)(x0 + 1) < TBL_)       v01 = rpe[(y0 * TBL_ + x0 + 1) * NH_ + h];
      if ((unsigned)(y0 + 1) < TBL_ && (unsigned)x0 < TBL_)       v10 = rpe[((y0 + 1) * TBL_ + x0) * NH_ + h];
      if ((unsigned)(y0 + 1) < TBL_ && (unsigned)(x0 + 1) < TBL_) v11 = rpe[((y0 + 1) * TBL_ + x0 + 1) * NH_ + h];
      const float bias = v00 * (1 - wy) * (1 - wx) + v01 * (1 - wy) * wx
                       + v10 * wy * (1 - wx)       + v11 * wy * wx;
      const float sv = s[rr] * scale + bias;
      p[rr] = sv;
      tmax = fmaxf(tmax, sv);
    }
    tmax = fmaxf(tmax, __shfl_xor(tmax, 16, 32));   // combine key halves

    // ---- online softmax update ----
    const float mnew  = fmaxf(mrow, tmax);
    const float alpha = __expf(mrow - mnew);
    float tsum = 0.f;
    #pragma unroll
    for (int rr = 0; rr < 8; ++rr) { p[rr] = __expf(p[rr] - mnew); tsum += p[rr]; }
    tsum += __shfl_xor(tsum, 16, 32);
    lrow = lrow * alpha + tsum;
    mrow = mnew;
    #pragma unroll
    for (int rr = 0; rr < 8; ++rr) { o0[rr] *= alpha; o1[rr] *= alpha; }

    // ---- O^T += V^T @ P^T ----
    #pragma unroll
    for (int c = 0; c < 4; ++c) {
      const int rb = (c & 1) * 4;
      const int sh = c >> 1;                // source half holding keys of this chunk
      const float q0 = p[rb], q1 = p[rb + 1], q2 = p[rb + 2], q3 = p[rb + 3];
      const float t0 = __shfl_xor(q0, 16, 32), t1 = __shfl_xor(q1, 16, 32);
      const float t2 = __shfl_xor(q2, 16, 32), t3 = __shfl_xor(q3, 16, 32);
      v2f bf;
      if (sh == 0) { bf.x = hl ? t2 : q0; bf.y = hl ? t3 : q1; }
      else         { bf.x = hl ? q2 : t0; bf.y = hl ? q3 : t1; }
      const int nk = kt * 16 + 4 * c + 2 * hl;
      v2f a0, a1;
      a0.x = Vp[(size_t)nk * NC_ + h * DH_ + r];
      a0.y = Vp[(size_t)(nk + 1) * NC_ + h * DH_ + r];
      a1.x = Vp[(size_t)nk * NC_ + h * DH_ + 16 + r];
      a1.y = Vp[(size_t)(nk + 1) * NC_ + h * DH_ + 16 + r];
      o0 = __builtin_amdgcn_wmma_f32_16x16x4_f32(false, a0, false, bf, (short)0, o0, false, false);
      o1 = __builtin_amdgcn_wmma_f32_16x16x4_f32(false, a1, false, bf, (short)0, o1, false, false);
    }
  }

  // ---- normalize + store O (out[b][query][h*32+d]) ----
  const float inv = 1.0f / lrow;
  float* op = outb + ((size_t)b * NS_ + nq) * NC_ + h * DH_;
  #pragma unroll
  for (int rr = 0; rr < 8; ++rr) {
    op[rr + 8 * hl]      = o0[rr] * inv;
    op[16 + rr + 8 * hl] = o1[rr] * inv;
  }
}

// ---------------------------------------------------------------------------
extern "C" void kernel_launch(void* const* d_in, const int* in_sizes, int n_in,
                              void* d_out, int out_size, void* d_ws, size_t ws_size,
                              hipStream_t stream) {
  (void)in_sizes; (void)n_in; (void)out_size; (void)ws_size;
  const float* x      = (const float*)d_in[0];
  const float* w_off  = (const float*)d_in[1];
  const float* b_off  = (const float*)d_in[2];
  const float* ln_g   = (const float*)d_in[3];
  const float* ln_b   = (const float*)d_in[4];
  const float* w_offp = (const float*)d_in[5];
  const float* wq     = (const float*)d_in[6];
  const float* bq     = (const float*)d_in[7];
  const float* wk     = (const float*)d_in[8];
  const float* bk     = (const float*)d_in[9];
  const float* wv     = (const float*)d_in[10];
  const float* bv     = (const float*)d_in[11];
  const float* wo     = (const float*)d_in[12];
  const float* bo     = (const float*)d_in[13];
  const float* rpe    = (const float*)d_in[14];

  float* y       = (float*)d_out;                 // [2,32,32,384] = 786432
  float* pos_out = y + (size_t)B_ * NS_ * NC_;    // [2,6,32,32,2] = 24576
  float* ref_out = pos_out + (size_t)B_ * NG_ * NS_ * 2;

  float* ws      = (float*)d_ws;
  const size_t SZ = (size_t)B_ * NS_ * NC_;       // 786432
  float* qbuf    = ws;
  float* kbuf    = qbuf + SZ;
  float* vbuf    = kbuf + SZ;
  float* convbuf = vbuf + SZ;
  float* attnout = convbuf + SZ;

  const int M = B_ * NS_, N = NC_, K = NC_;
  const dim3 blk(256);
  const int gemmBlocks = (M / 16) * (N / 64) / 8;          // 96

  gemm_bias<<<gemmBlocks, blk, 0, stream>>>(x, wq, bq, qbuf, M, N, K);
  gemm_bias<<<gemmBlocks, blk, 0, stream>>>(x, wk, bk, kbuf, M, N, K);
  gemm_bias<<<gemmBlocks, blk, 0, stream>>>(x, wv, bv, vbuf, M, N, K);

  conv_off<<<(B_ * NS_ * NC_) / 256, blk, 0, stream>>>(qbuf, w_off, b_off, convbuf);
  ln_offset<<<(B_ * NS_) / 8, blk, 0, stream>>>(convbuf, ln_g, ln_b, w_offp, pos_out, ref_out);

  attn_fused<<<(24 * 64) / 8, blk, 0, stream>>>(qbuf, kbuf, vbuf, pos_out, rpe, attnout);

  gemm_bias<<<gemmBlocks, blk, 0, stream>>>(attnout, wo, bo, y, M, N, K);
}